// CoCLR_8074538517166
// MI455X (gfx1250) — compile-verified
//
#include <hip/hip_runtime.h>
#include <hip/hip_bf16.h>

// Problem constants (from reference): B=32, D=128, M=65536
#define B_    32
#define D_    128
#define M_    65536
#define K0_   16384u   // stage-0 keep count (M * 0.25)
#define K1_   4096u    // stage-1 keep count (M * 0.0625)
#define TOPK_ 16u

typedef float v2f __attribute__((ext_vector_type(2)));
typedef float v8f __attribute__((ext_vector_type(8)));

// Monotonic float -> uint map: order-preserving for all finite + inf values.
__device__ __forceinline__ unsigned fkey(float f) {
  unsigned u = __float_as_uint(f);
  return (u & 0x80000000u) ? ~u : (u | 0x80000000u);
}

// ---------------------------------------------------------------------------
// Kernel 1: S_main = anchor_main @ bank_main, S_aux = anchor_aux @ bank_aux
// via V_WMMA_F32_16X16X4_F32. One wave per 16-column tile computes the FULL
// 32x16 output (both batch row-tiles), so each bank element is loaded exactly
// once: compulsory traffic = 64 MB. 4 accumulators (main/aux x rt0/rt1),
// 128 WMMAs per wave. Bandwidth-bound; fp32 keeps top-k exact vs reference.
// ---------------------------------------------------------------------------
__global__ void __launch_bounds__(256)
sim_gemm_wmma(const float* __restrict__ anchor_main,
              const float* __restrict__ anchor_aux,
              const float* __restrict__ bank_main,   // [D, M] row-major
              const float* __restrict__ bank_aux,    // [D, M] row-major
              const unsigned char* __restrict__ mask, // [B, M] jax bool (1B)
              float* __restrict__ sim_main,           // [B, M]
              float* __restrict__ sim_aux)            // [B, M]
{
  const int gtid  = blockIdx.x * blockDim.x + threadIdx.x;
  const int wave  = gtid >> 5;          // 0 .. 4095 : column tile id
  const int lane  = threadIdx.x & 31;
  const int n0    = wave << 4;
  const int rlo   = lane & 15;          // tile row (A/M) or tile col (B/C/D)
  const int khalf = lane >> 4;          // 0: K in {k,k+1}; 1: K in {k+2,k+3}

  // A-matrix 16x4 f32: lane L<16 holds (M=L, K=k..k+1), L>=16 -> K=k+2..k+3
  const float* am0 = anchor_main + (rlo)      * D_ + khalf * 2;  // rows 0..15
  const float* am1 = anchor_main + (16 + rlo) * D_ + khalf * 2;  // rows 16..31
  const float* aa0 = anchor_aux  + (rlo)      * D_ + khalf * 2;
  const float* aa1 = anchor_aux  + (16 + rlo) * D_ + khalf * 2;
  // B-matrix 4x16 f32: VGPR j, lane L<16 holds B[j][L]; L>=16 holds B[j+2][L-16]
  const size_t coff = (size_t)(khalf * 2) * (size_t)M_ + (size_t)(n0 + rlo);
  const float* bm = bank_main + coff;
  const float* ba = bank_aux  + coff;

  v8f accm0 = {}, accm1 = {}, acca0 = {}, acca1 = {};
  for (int k = 0; k < D_; k += 4) {
    if (k + 8 < D_) {
      __builtin_prefetch(bm + (size_t)(k + 8) * (size_t)M_, 0, 1);
      __builtin_prefetch(ba + (size_t)(k + 8) * (size_t)M_, 0, 1);
    }
    // Shared B tiles (loaded once, used for both row-tiles)
    v2f Bm = { bm[(size_t)k * (size_t)M_], bm[(size_t)(k + 1) * (size_t)M_] };
    v2f Ba = { ba[(size_t)k * (size_t)M_], ba[(size_t)(k + 1) * (size_t)M_] };
    v2f Am0 = { am0[k], am0[k + 1] };
    v2f Am1 = { am1[k], am1[k + 1] };
    v2f Aa0 = { aa0[k], aa0[k + 1] };
    v2f Aa1 = { aa1[k], aa1[k + 1] };
    // 8 args: (neg_a, A, neg_b, B, c_mod, C, reuse_a, reuse_b)
    accm0 = __builtin_amdgcn_wmma_f32_16x16x4_f32(false, Am0, false, Bm,
                                                  (short)0, accm0, false, false);
    accm1 = __builtin_amdgcn_wmma_f32_16x16x4_f32(false, Am1, false, Bm,
                                                  (short)0, accm1, false, false);
    acca0 = __builtin_amdgcn_wmma_f32_16x16x4_f32(false, Aa0, false, Ba,
                                                  (short)0, acca0, false, false);
    acca1 = __builtin_amdgcn_wmma_f32_16x16x4_f32(false, Aa1, false, Ba,
                                                  (short)0, acca1, false, false);
  }

  // C/D layout: lane<16 -> (M=v, N=lane); lane>=16 -> (M=v+8, N=lane-16)
  const int col = n0 + rlo;
  #pragma unroll
  for (int rt = 0; rt < 2; ++rt) {
    const v8f accm = rt ? accm1 : accm0;
    const v8f acca = rt ? acca1 : acca0;
    #pragma unroll
    for (int v = 0; v < 8; ++v) {
      const int brow = rt * 16 + khalf * 8 + v;
      const size_t off = (size_t)brow * (size_t)M_ + (size_t)col;
      float s = accm[v];
      if (mask[off]) s = -__builtin_inff();  // reference masks only S_main
      sim_main[off] = s;
      sim_aux[off]  = acca[v];
    }
  }
}

// ---------------------------------------------------------------------------
// MSB-first 8-bit radix select of the k-th largest key among {m : pred(m)}.
// Returns the exact threshold key value. One workgroup; hist in LDS.
// ---------------------------------------------------------------------------
template <typename Pred>
__device__ unsigned radix_select_desc(const float* __restrict__ keys, int n,
                                      unsigned k, Pred pred,
                                      unsigned* hist, unsigned* sh,
                                      int tid, int nt)
{
  unsigned prefix = 0, kk = k;
  for (int shift = 24; shift >= 0; shift -= 8) {
    for (int i = tid; i < 256; i += nt) hist[i] = 0u;
    __syncthreads();
    const unsigned prefMask = (shift == 24) ? 0u : (0xFFFFFFFFu << (shift + 8));
    for (int m = tid; m < n; m += nt) {
      if (!pred(m)) continue;
      const unsigned u = fkey(keys[m]);
      if ((u & prefMask) == prefix)
        atomicAdd(&hist[(u >> shift) & 255u], 1u);
    }
    __syncthreads();
    if (tid == 0) {
      unsigned acc = 0;
      int d = 255;
      for (;; --d) {
        const unsigned c = hist[d];
        if (acc + c >= kk || d == 0) { sh[0] = (unsigned)d; sh[1] = kk - acc; break; }
        acc += c;
      }
    }
    __syncthreads();
    prefix |= sh[0] << shift;
    kk = sh[1];
    __syncthreads();
  }
  return prefix;
}

// ---------------------------------------------------------------------------
// Kernel 2: one workgroup per batch row. Thresholds:
//   tm = 16384th largest S_main              (stage 0)
//   ta = 4096th  largest S_aux  | main>=tm   (stage 1)
//   tf = 16th    largest S_main | main>=tm && aux>=ta  (stages 2+3 fused:
//        top-16-by-main of top-1024-by-main == top-16-by-main of the set)
// Then gather candidates and emit 16 indices sorted by (value desc, idx asc).
// All score reads are L2-resident (16 MB << 192 MB L2).
// ---------------------------------------------------------------------------
__global__ void __launch_bounds__(256)
cascade_topk(const float* __restrict__ sim_main,
             const float* __restrict__ sim_aux,
             float* __restrict__ out)   // [B*TOPK indices][B*TOPK weights]
{
  const int row = blockIdx.x;
  const int tid = threadIdx.x, nt = blockDim.x;
  const float* vm = sim_main + (size_t)row * (size_t)M_;
  const float* va = sim_aux  + (size_t)row * (size_t)M_;

  __shared__ unsigned hist[256];
  __shared__ unsigned sh[2];
  __shared__ unsigned cand_cnt;
  __shared__ unsigned cand_key[64];
  __shared__ int      cand_idx[64];

  const unsigned tm = radix_select_desc(vm, M_, K0_,
      [&](int) { return true; }, hist, sh, tid, nt);
  const unsigned ta = radix_select_desc(va, M_, K1_,
      [&](int m) { return fkey(vm[m]) >= tm; }, hist, sh, tid, nt);
  const unsigned tf = radix_select_desc(vm, M_, TOPK_,
      [&](int m) { return fkey(vm[m]) >= tm && fkey(va[m]) >= ta; },
      hist, sh, tid, nt);

  if (tid == 0) cand_cnt = 0u;
  __syncthreads();
  for (int m = tid; m < M_; m += nt) {
    const unsigned um = fkey(vm[m]);      // um >= tf implies um >= tm
    if (um >= tf && fkey(va[m]) >= ta) {
      const unsigned p = atomicAdd(&cand_cnt, 1u);
      if (p < 64u) { cand_key[p] = um; cand_idx[p] = m; }
    }
  }
  __syncthreads();

  if (tid == 0) {
    const int cnt = (int)(cand_cnt < 64u ? cand_cnt : 64u);
    for (int j = 0; j < (int)TOPK_; ++j) {
      if (j < cnt) {
        int best = j;
        for (int i = j + 1; i < cnt; ++i) {
          if (cand_key[i] > cand_key[best] ||
              (cand_key[i] == cand_key[best] && cand_idx[i] < cand_idx[best]))
            best = i;
        }
        const unsigned bk = cand_key[best]; const int bi = cand_idx[best];
        cand_key[best] = cand_key[j]; cand_idx[best] = cand_idx[j];
        cand_key[j] = bk;             cand_idx[j]  = bi;
        out[row * (int)TOPK_ + j] = (float)bi;
      } else {
        out[row * (int)TOPK_ + j] = 0.0f;
      }
      out[B_ * (int)TOPK_ + row * (int)TOPK_ + j] = 1.0f;  // pos_weights = 1
    }
  }
}

// ---------------------------------------------------------------------------
extern "C" void kernel_launch(void* const* d_in, const int* in_sizes, int n_in,
                              void* d_out, int out_size, void* d_ws, size_t ws_size,
                              hipStream_t stream) {
  (void)in_sizes; (void)n_in; (void)out_size; (void)ws_size;
  const float* anchor_main = (const float*)d_in[0];          // [B, D]
  const float* anchor_aux  = (const float*)d_in[1];          // [B, D]
  const float* bank_main   = (const float*)d_in[2];          // [D, M]
  const float* bank_aux    = (const float*)d_in[3];          // [D, M]
  // d_in[4] = index_record (arange[M]) -> identity mapping, not needed
  const unsigned char* mask = (const unsigned char*)d_in[5]; // [B, M] bool

  float* sim_main = (float*)d_ws;                            // 8 MB
  float* sim_aux  = sim_main + (size_t)B_ * (size_t)M_;      // 8 MB

  // 4096 column tiles, each wave computes the full 32x16 output strip:
  // 512 blocks * 8 wave32 = 4096 waves; banks streamed exactly once (64 MB).
  sim_gemm_wmma<<<dim3(512), dim3(256), 0, stream>>>(
      anchor_main, anchor_aux, bank_main, bank_aux, mask, sim_main, sim_aux);

  cascade_topk<<<dim3(B_), dim3(256), 0, stream>>>(
      sim_main, sim_aux, (float*)d_out);
}